// FCAM_Module_69389491634868
// MI455X (gfx1250) — compile-verified
//
#include <hip/hip_runtime.h>

// ---------------------------------------------------------------------------
// FCAM channel-attention module for MI455X (gfx1250, wave32)
//   - f32 WMMA 16x16x4 for all GEMMs (reference is f32; problem is HBM-bound)
//   - GLOBAL_LOAD_ASYNC_TO_LDS_B128 double buffering for the two kernels that
//     stream big tensors (rgb conv, final W2@Z GEMM)
// ---------------------------------------------------------------------------

typedef __attribute__((ext_vector_type(2))) float v2f;
typedef __attribute__((ext_vector_type(8))) float v8f;

#define BATCH 32
#define CCH   256      // channels C
#define CQ    32       // C/8
#define NT    1024     // t-branch H*W (32*32)
#define NS    4096     // s-branch H*W (64*64)

__device__ __forceinline__ v8f wmma4(v2f a, v2f b, v8f c) {
    // D = A(16x4) * B(4x16) + C(16x16), all f32
    return __builtin_amdgcn_wmma_f32_16x16x4_f32(false, a, false, b,
                                                 (short)0, c, false, false);
}

// async DMA: 16 bytes global -> LDS, tracked by ASYNCcnt (no VGPR data path)
__device__ __forceinline__ void async_b128(void* lds_ptr, const void* gptr) {
    const unsigned lds_off = (unsigned)(unsigned long long)lds_ptr; // flat low32 == LDS offset
    asm volatile("global_load_async_to_lds_b128 %0, %1, off"
                 :: "v"(lds_off), "v"(gptr)
                 : "memory");
}
__device__ __forceinline__ void wait_async0() {
    asm volatile("s_wait_asynccnt 0" ::: "memory");
}

// ---------------------------------------------------------------------------
// Kernel 1: fused Q/K 1x1 conv.  Q = qw @ X + qb, K = kw @ X + kb
// X = rgb[b] as [C, N].  Output [B, CQ, N] (== [B, C, N/8] reinterpreted).
// grid (N/64, B), block 256 (8 waves, 2x4 wave grid over a 32x64 tile)
// X tile staged by async-to-LDS, double buffered.
// ---------------------------------------------------------------------------
__global__ __launch_bounds__(256)
void conv_qk_kernel(const float* __restrict__ X,
                    const float* __restrict__ qw, const float* __restrict__ qb,
                    const float* __restrict__ kw, const float* __restrict__ kb,
                    float* __restrict__ Qout, float* __restrict__ Kout, int N)
{
    const int b    = blockIdx.y;
    const int n0   = blockIdx.x * 64;
    const int wave = threadIdx.x >> 5;
    const int lane = threadIdx.x & 31;
    const int wr   = wave >> 2;        // 0..1 : 16-row sub-tile
    const int wc   = wave & 3;         // 0..3 : 16-col sub-tile
    const int half = lane >> 4;        // K-half select for f32 frags
    const int mn   = lane & 15;
    const int m0   = wr * 16;
    const int nn   = n0 + wc * 16 + mn;

    const float* Xb = X + (size_t)b * CCH * N;

    // pad 72 floats (288B): 16B-aligned rows; half-wave bank sets disjoint
    __shared__ __align__(16) float Bs[2][16][72];

    // staging coords: 16x64 tile = 256 x 16B chunks, one per thread
    const int skk = threadIdx.x >> 4;          // 0..15
    const int sj  = (threadIdx.x & 15) * 4;    // 0..60

    // prefetch k-block 0 into buffer 0
    async_b128(&Bs[0][skk][sj], Xb + (size_t)skk * N + n0 + sj);

    v8f accq = {}; v8f acck = {};
    for (int kb2 = 0; kb2 < CCH / 16; ++kb2) {
        const int cur = kb2 & 1;
        wait_async0();
        __syncthreads();
        if (kb2 + 1 < CCH / 16) {
            const int k0n = (kb2 + 1) * 16;
            async_b128(&Bs[1 - cur][skk][sj], Xb + (size_t)(k0n + skk) * N + n0 + sj);
        }
        const int k0 = kb2 * 16;
        #pragma unroll
        for (int kk = 0; kk < 16; kk += 4) {
            const int kx = k0 + kk + half * 2;     // global k for A frags
            const int kl = kk + half * 2;          // LDS k for B frag
            v2f aq, ak, bb;
            const float* qa = qw + (size_t)(m0 + mn) * CCH + kx;
            const float* ka = kw + (size_t)(m0 + mn) * CCH + kx;
            aq.x = qa[0]; aq.y = qa[1];
            ak.x = ka[0]; ak.y = ka[1];
            bb.x = Bs[cur][kl][wc * 16 + mn];
            bb.y = Bs[cur][kl + 1][wc * 16 + mn];
            accq = wmma4(aq, bb, accq);
            acck = wmma4(ak, bb, acck);
        }
    }
    float* Qb = Qout + (size_t)b * CQ * N;
    float* Kb = Kout + (size_t)b * CQ * N;
    #pragma unroll
    for (int r = 0; r < 8; ++r) {
        const int m = m0 + r + half * 8;
        Qb[(size_t)m * N + nn] = accq[r] + qb[m];
        Kb[(size_t)m * N + nn] = acck[r] + kb[m];
    }
}

// ---------------------------------------------------------------------------
// Kernel 2a: energy[b] = Qr @ Kr^T, Qr/Kr = [C, Mdim] views (Mdim = N/8)
// grid (32, B), block 256; each wave one 16x16 tile of the 256x256 output
// ---------------------------------------------------------------------------
__global__ __launch_bounds__(256)
void energy_kernel(const float* __restrict__ Q, const float* __restrict__ K,
                   float* __restrict__ E, int Mdim)
{
    const int b    = blockIdx.y;
    const int wave = threadIdx.x >> 5;
    const int lane = threadIdx.x & 31;
    const int tile = blockIdx.x * 8 + wave;      // 0..255
    const int tr   = tile >> 4, tc = tile & 15;
    const int half = lane >> 4, mn = lane & 15;

    const float* Qr = Q + (size_t)b * CCH * Mdim + (size_t)(tr * 16 + mn) * Mdim;
    const float* Kr = K + (size_t)b * CCH * Mdim + (size_t)(tc * 16 + mn) * Mdim;

    v8f acc = {};
    for (int k = 0; k < Mdim; k += 4) {
        const int kx = k + half * 2;
        v2f a, bb;
        a.x  = Qr[kx]; a.y  = Qr[kx + 1];        // A[m][k]
        bb.x = Kr[kx]; bb.y = Kr[kx + 1];        // B[k][n] = Kr[n][k]
        acc = wmma4(a, bb, acc);
    }
    float* Eb = E + (size_t)b * CCH * CCH;
    #pragma unroll
    for (int r = 0; r < 8; ++r)
        Eb[(size_t)(tr * 16 + r + half * 8) * CCH + tc * 16 + mn] = acc[r];
}

// ---------------------------------------------------------------------------
// Kernel 2b: reversed softmax, in place.
// softmax(rowmax - e) == exp(rowmin - e) / sum(exp(rowmin - e))
// one wave per 256-wide row
// ---------------------------------------------------------------------------
__global__ __launch_bounds__(256)
void softmax_rev_kernel(float* __restrict__ E)
{
    const int row  = blockIdx.x * 8 + (threadIdx.x >> 5);
    const int lane = threadIdx.x & 31;
    float* e = E + (size_t)row * CCH;

    float v[8];
    float mn = 3.402823466e38f;
    #pragma unroll
    for (int i = 0; i < 8; ++i) { v[i] = e[lane + i * 32]; mn = fminf(mn, v[i]); }
    #pragma unroll
    for (int off = 16; off; off >>= 1) mn = fminf(mn, __shfl_xor(mn, off, 32));
    float s = 0.f;
    #pragma unroll
    for (int i = 0; i < 8; ++i) { v[i] = __expf(mn - v[i]); s += v[i]; }
    #pragma unroll
    for (int off = 16; off; off >>= 1) s += __shfl_xor(s, off, 32);
    const float inv = 1.0f / s;
    #pragma unroll
    for (int i = 0; i < 8; ++i) e[lane + i * 32] = v[i] * inv;
}

// ---------------------------------------------------------------------------
// Kernel 3: W2[b] = attn[b] @ Vw   (fold attention into value weights)
// grid (32, B), block 256; wave-per-tile like energy_kernel
// ---------------------------------------------------------------------------
__global__ __launch_bounds__(256)
void w2_kernel(const float* __restrict__ Attn, const float* __restrict__ Vw,
               float* __restrict__ W2)
{
    const int b    = blockIdx.y;
    const int wave = threadIdx.x >> 5;
    const int lane = threadIdx.x & 31;
    const int tile = blockIdx.x * 8 + wave;
    const int tr   = tile >> 4, tc = tile & 15;
    const int half = lane >> 4, mn = lane & 15;

    const float* Ab = Attn + (size_t)b * CCH * CCH + (size_t)(tr * 16 + mn) * CCH;

    v8f acc = {};
    for (int k = 0; k < CCH; k += 4) {
        const int kx = k + half * 2;
        v2f a, bb;
        a.x  = Ab[kx]; a.y = Ab[kx + 1];
        bb.x = Vw[(size_t)kx * CCH + tc * 16 + mn];
        bb.y = Vw[(size_t)(kx + 1) * CCH + tc * 16 + mn];
        acc = wmma4(a, bb, acc);
    }
    float* Ob = W2 + (size_t)b * CCH * CCH;
    #pragma unroll
    for (int r = 0; r < 8; ++r)
        Ob[(size_t)(tr * 16 + r + half * 8) * CCH + tc * 16 + mn] = acc[r];
}

// ---------------------------------------------------------------------------
// Kernel 3b: b2[b][c] = attn[b][c][:] . vb  (tiny)
// ---------------------------------------------------------------------------
__global__ __launch_bounds__(256)
void bias2_kernel(const float* __restrict__ Attn, const float* __restrict__ vb,
                  float* __restrict__ b2)
{
    const int b = blockIdx.x, c = threadIdx.x;
    const float* a = Attn + ((size_t)b * CCH + c) * CCH;
    float s = 0.f;
    for (int d = 0; d < CCH; ++d) s += a[d] * vb[d];
    b2[b * CCH + c] = s;
}

// ---------------------------------------------------------------------------
// Kernel 4: out = alpha * (W2 @ Z + b2) + Z      (the only big-tensor GEMM)
// grid (N/32, 4, B), block 256; tile 64x32, kb=16
// Both tiles staged by async-to-LDS, double buffered.
// ---------------------------------------------------------------------------
__global__ __launch_bounds__(256)
void out_kernel(const float* __restrict__ W2, const float* __restrict__ b2,
                const float* __restrict__ Z, const float* __restrict__ scale,
                float* __restrict__ Out, int N)
{
    const int b    = blockIdx.z;
    const int c0   = blockIdx.y * 64;
    const int n0   = blockIdx.x * 32;
    const int wave = threadIdx.x >> 5;
    const int lane = threadIdx.x & 31;
    const int wr   = wave >> 1;      // 0..3
    const int wc   = wave & 1;       // 0..1
    const int half = lane >> 4, mn = lane & 15;

    // pads: 20 floats (80B) and 40 floats (160B): rows stay 16B-aligned for
    // async B128 stores, bank sets conflict-free for the WMMA fragment reads
    __shared__ __align__(16) float As[2][64][20];   // W2 tile
    __shared__ __align__(16) float Bs[2][16][40];   // Z tile

    const float* W2b = W2 + (size_t)b * CCH * CCH;
    const float* Zb  = Z  + (size_t)b * CCH * N;

    // staging coords: A tile 64x16 = 256 chunks (all threads),
    //                 B tile 16x32 = 128 chunks (threads 0..127)
    const int ai  = threadIdx.x >> 2;          // 0..63
    const int ak  = (threadIdx.x & 3) * 4;     // 0,4,8,12
    const int bkz = threadIdx.x >> 3;          // 0..31 (valid < 16)
    const int bj  = (threadIdx.x & 7) * 4;     // 0..28
    const bool doB = (threadIdx.x < 128);

    // prefetch k-block 0 into buffer 0
    async_b128(&As[0][ai][ak], W2b + (size_t)(c0 + ai) * CCH + ak);
    if (doB) async_b128(&Bs[0][bkz][bj], Zb + (size_t)bkz * N + n0 + bj);

    v8f acc = {};
    for (int kb2 = 0; kb2 < CCH / 16; ++kb2) {
        const int cur = kb2 & 1;
        wait_async0();          // my chunks of tile kb2 have landed in LDS
        __syncthreads();        // everyone's chunks have landed
        if (kb2 + 1 < CCH / 16) {
            const int k0n = (kb2 + 1) * 16;
            async_b128(&As[1 - cur][ai][ak], W2b + (size_t)(c0 + ai) * CCH + k0n + ak);
            if (doB) async_b128(&Bs[1 - cur][bkz][bj], Zb + (size_t)(k0n + bkz) * N + n0 + bj);
        }
        #pragma unroll
        for (int kk = 0; kk < 16; kk += 4) {
            const int kl = kk + half * 2;
            v2f a, bb;
            a.x  = As[cur][wr * 16 + mn][kl];
            a.y  = As[cur][wr * 16 + mn][kl + 1];
            bb.x = Bs[cur][kl][wc * 16 + mn];
            bb.y = Bs[cur][kl + 1][wc * 16 + mn];
            acc = wmma4(a, bb, acc);
        }
    }
    const float alpha = scale[0];
    float* Ob = Out + (size_t)b * CCH * N;
    const int nn = n0 + wc * 16 + mn;
    #pragma unroll
    for (int r = 0; r < 8; ++r) {
        const int c = c0 + wr * 16 + r + half * 8;
        const float z = Zb[(size_t)c * N + nn];
        Ob[(size_t)c * N + nn] = alpha * (acc[r] + b2[b * CCH + c]) + z;
    }
}

// ---------------------------------------------------------------------------
// Host launcher
// ---------------------------------------------------------------------------
extern "C" void kernel_launch(void* const* d_in, const int* in_sizes, int n_in,
                              void* d_out, int out_size, void* d_ws, size_t ws_size,
                              hipStream_t stream)
{
    (void)in_sizes; (void)n_in; (void)out_size; (void)ws_size;

    const float* rgb_t = (const float*)d_in[0];
    const float* rgb_s = (const float*)d_in[1];
    const float* z_t   = (const float*)d_in[2];
    const float* z_s   = (const float*)d_in[3];
    const float* tq_w  = (const float*)d_in[4];
    const float* tq_b  = (const float*)d_in[5];
    const float* tk_w  = (const float*)d_in[6];
    const float* tk_b  = (const float*)d_in[7];
    const float* tv_w  = (const float*)d_in[8];
    const float* tv_b  = (const float*)d_in[9];
    const float* sq_w  = (const float*)d_in[10];
    const float* sq_b  = (const float*)d_in[11];
    const float* sk_w  = (const float*)d_in[12];
    const float* sk_b  = (const float*)d_in[13];
    const float* sv_w  = (const float*)d_in[14];
    const float* sv_b  = (const float*)d_in[15];
    const float* alpha = (const float*)d_in[16];
    const float* beta  = (const float*)d_in[17];

    float* out_t = (float*)d_out;
    float* out_s = out_t + (size_t)BATCH * CCH * NT;

    // workspace carve-up (floats)
    float* ws = (float*)d_ws;
    size_t off = 0;
    float* q_t    = ws + off; off += (size_t)BATCH * CQ * NT;
    float* k_t    = ws + off; off += (size_t)BATCH * CQ * NT;
    float* q_s    = ws + off; off += (size_t)BATCH * CQ * NS;
    float* k_s    = ws + off; off += (size_t)BATCH * CQ * NS;
    float* attn_t = ws + off; off += (size_t)BATCH * CCH * CCH;
    float* attn_s = ws + off; off += (size_t)BATCH * CCH * CCH;
    float* w2_t   = ws + off; off += (size_t)BATCH * CCH * CCH;
    float* w2_s   = ws + off; off += (size_t)BATCH * CCH * CCH;
    float* b2_t   = ws + off; off += (size_t)BATCH * CCH;
    float* b2_s   = ws + off; off += (size_t)BATCH * CCH;

    // ---------------- t branch ----------------
    conv_qk_kernel<<<dim3(NT / 64, BATCH), 256, 0, stream>>>(rgb_t, tq_w, tq_b, tk_w, tk_b,
                                                             q_t, k_t, NT);
    energy_kernel<<<dim3(32, BATCH), 256, 0, stream>>>(q_t, k_t, attn_t, NT / 8);
    softmax_rev_kernel<<<BATCH * CCH / 8, 256, 0, stream>>>(attn_t);
    w2_kernel<<<dim3(32, BATCH), 256, 0, stream>>>(attn_t, tv_w, w2_t);
    bias2_kernel<<<BATCH, CCH, 0, stream>>>(attn_t, tv_b, b2_t);
    out_kernel<<<dim3(NT / 32, 4, BATCH), 256, 0, stream>>>(w2_t, b2_t, z_t, alpha, out_t, NT);

    // ---------------- s branch ----------------
    conv_qk_kernel<<<dim3(NS / 64, BATCH), 256, 0, stream>>>(rgb_s, sq_w, sq_b, sk_w, sk_b,
                                                             q_s, k_s, NS);
    energy_kernel<<<dim3(32, BATCH), 256, 0, stream>>>(q_s, k_s, attn_s, NS / 8);
    softmax_rev_kernel<<<BATCH * CCH / 8, 256, 0, stream>>>(attn_s);
    w2_kernel<<<dim3(32, BATCH), 256, 0, stream>>>(attn_s, sv_w, w2_s);
    bias2_kernel<<<BATCH, CCH, 0, stream>>>(attn_s, sv_b, b2_s);
    out_kernel<<<dim3(NS / 32, 4, BATCH), 256, 0, stream>>>(w2_s, b2_s, z_s, beta, out_s, NS);
}